// MaxMinAgg_27754078667037
// MI455X (gfx1250) — compile-verified
//
#include <hip/hip_runtime.h>
#include <stdint.h>

// Problem constants (from reference): B=1024, IN_F=256, OUT_F=128, AGG=4
#define B_ROWS 1024
#define IN_F   256
#define OUT_F  128
#define AGG    4
#define ROWS_PER_BLOCK 8   // rows of m handled per workgroup

typedef __attribute__((ext_vector_type(4))) unsigned int v4u;
typedef __attribute__((ext_vector_type(8))) int          v8i;
typedef __attribute__((ext_vector_type(4))) int          v4i;

// ---------------------------------------------------------------------------
// Kernel A: W'[k][o] = max over 4 consecutive cols of weight[k][4o..4o+3].
// weight is [256][512] row-major; one float4 per thread, fully coalesced.
// Valid because max_a min(x, w_a) == min(x, max_a w_a)  (min is monotone).
// ---------------------------------------------------------------------------
__global__ __launch_bounds__(256)
void reduce_weight_kernel(const float* __restrict__ w, float* __restrict__ wp) {
    int g = blockIdx.x * 256 + threadIdx.x;          // 0 .. 32767
    const float4* w4 = (const float4*)w;             // w4[k*128 + o] = w[k][4o..4o+3]
    float4 v = w4[g];
    wp[g] = fmaxf(fmaxf(v.x, v.y), fmaxf(v.z, v.w)); // W'[k][o], row-major [256][128]
}

// ---------------------------------------------------------------------------
// Kernel B: out[b,o] = max_k min(m[b,k], W'[k,o])
//  - whole W' (128 KB) pulled into LDS by ONE tensor_load_to_lds (TDM, wave 0)
//  - this block's 8 m-rows (8 KB) pulled by global_load_async_to_lds_b128
//  - 256 threads: o = tid&127, half = tid>>7 -> 4 rows each, k-loop by 4
// ---------------------------------------------------------------------------
__global__ __launch_bounds__(256)
void maxmin_kernel(const float* __restrict__ m, const float* __restrict__ wp,
                   float* __restrict__ out) {
    extern __shared__ float smem[];
    float* lds_w = smem;                   // [256][128] floats = 128 KB
    float* lds_m = smem + IN_F * OUT_F;    // [8][256]  floats =   8 KB

    const int tid     = threadIdx.x;
    const int rowBase = blockIdx.x * ROWS_PER_BLOCK;

    // ---- TDM bulk load of W' into LDS (one DMA descriptor, issued by wave 0)
    if (tid < 32) {
        unsigned lds_off = (unsigned)(uintptr_t)lds_w;
        unsigned long long ga = (unsigned long long)(uintptr_t)wp;
        // D# group0 (128b): count=1 | lds_addr | global_addr[56:0] | type=2
        v4u g0 = { 1u,
                   lds_off,
                   (unsigned)(ga & 0xffffffffull),
                   (unsigned)((ga >> 32) & 0x1ffffffull) | (2u << 30) };
        // D# group1 (256b): data_size=2 (4B); tensor_dim0=32768; tensor_dim1=1;
        // tile_dim0=32768; tile_dim1=1; tensor_dim0_stride=32768
        v8i g1 = { (int)(2u << 16),        // workgroup_mask=0, data_size=4B
                   (int)0x80000000u,       // tensor_dim0 low16 (32768) in [31:16]
                   (int)0x00010000u,       // tensor_dim0 hi=0, tensor_dim1=1
                   (int)0x80000000u,       // tensor_dim1 hi=0, tile_dim0=32768
                   1,                      // tile_dim1=1, tile_dim2=0
                   32768,                  // tensor_dim0_stride[31:0]
                   0, 0 };
        v4i g2 = {0,0,0,0};
        v4i g3 = {0,0,0,0};
#if defined(__clang_major__) && (__clang_major__ >= 23)
        v8i g4 = {0,0,0,0,0,0,0,0};
        __builtin_amdgcn_tensor_load_to_lds(g0, g1, g2, g3, g4, 0);
#else
        __builtin_amdgcn_tensor_load_to_lds(g0, g1, g2, g3, 0);
#endif
    }

    // ---- async copy of this block's 8 m-rows into LDS (ASYNCcnt path)
    {
        const float* msrc = m + (size_t)rowBase * IN_F;
        unsigned lm = (unsigned)(uintptr_t)lds_m;
        #pragma unroll
        for (int i = tid; i < (ROWS_PER_BLOCK * IN_F) / 4; i += 256) {
            unsigned loff = lm + (unsigned)i * 16u;   // LDS dest byte address
            unsigned voff = (unsigned)i * 16u;        // byte offset vs saddr
            asm volatile("global_load_async_to_lds_b128 %0, %1, %2"
                         :: "v"(loff), "v"(voff), "s"(msrc) : "memory");
        }
        asm volatile("s_wait_asynccnt 0x0" ::: "memory");
    }
    __builtin_amdgcn_s_wait_tensorcnt(0);
    __syncthreads();

    // ---- compute: each thread owns column o and 4 rows
    const int o    = tid & (OUT_F - 1);
    const int half = tid >> 7;                      // 0 or 1
    const float* mr0 = lds_m + (half * 4 + 0) * IN_F;
    const float* mr1 = lds_m + (half * 4 + 1) * IN_F;
    const float* mr2 = lds_m + (half * 4 + 2) * IN_F;
    const float* mr3 = lds_m + (half * 4 + 3) * IN_F;

    float acc0 = -__builtin_inff(), acc1 = -__builtin_inff();
    float acc2 = -__builtin_inff(), acc3 = -__builtin_inff();

    #pragma unroll 4
    for (int k = 0; k < IN_F; k += 4) {
        // lane-consecutive -> conflict-free ds_load_b32
        float w0 = lds_w[(k + 0) * OUT_F + o];
        float w1 = lds_w[(k + 1) * OUT_F + o];
        float w2 = lds_w[(k + 2) * OUT_F + o];
        float w3 = lds_w[(k + 3) * OUT_F + o];
        // wave-uniform broadcast ds_load_b128
        float4 a = *(const float4*)(mr0 + k);
        acc0 = fmaxf(fmaxf(acc0, fminf(a.x, w0)), fminf(a.y, w1)); // -> v_max3
        acc0 = fmaxf(fmaxf(acc0, fminf(a.z, w2)), fminf(a.w, w3));
        float4 b = *(const float4*)(mr1 + k);
        acc1 = fmaxf(fmaxf(acc1, fminf(b.x, w0)), fminf(b.y, w1));
        acc1 = fmaxf(fmaxf(acc1, fminf(b.z, w2)), fminf(b.w, w3));
        float4 c = *(const float4*)(mr2 + k);
        acc2 = fmaxf(fmaxf(acc2, fminf(c.x, w0)), fminf(c.y, w1));
        acc2 = fmaxf(fmaxf(acc2, fminf(c.z, w2)), fminf(c.w, w3));
        float4 d = *(const float4*)(mr3 + k);
        acc3 = fmaxf(fmaxf(acc3, fminf(d.x, w0)), fminf(d.y, w1));
        acc3 = fmaxf(fmaxf(acc3, fminf(d.z, w2)), fminf(d.w, w3));
    }

    float* orow = out + ((size_t)rowBase + half * 4) * OUT_F + o;
    orow[0 * OUT_F] = acc0;
    orow[1 * OUT_F] = acc1;
    orow[2 * OUT_F] = acc2;
    orow[3 * OUT_F] = acc3;
}

// ---------------------------------------------------------------------------
extern "C" void kernel_launch(void* const* d_in, const int* in_sizes, int n_in,
                              void* d_out, int out_size, void* d_ws, size_t ws_size,
                              hipStream_t stream) {
    const float* m  = (const float*)d_in[0];   // [1024,256] f32
    const float* w  = (const float*)d_in[1];   // [256,512]  f32
    float* out = (float*)d_out;                // [1024,128] f32
    float* wp  = (float*)d_ws;                 // scratch: W' [256,128] = 128 KB

    reduce_weight_kernel<<<(IN_F * OUT_F) / 256, 256, 0, stream>>>(w, wp);

    size_t shmem = (size_t)(IN_F * OUT_F + ROWS_PER_BLOCK * IN_F) * sizeof(float); // 136 KB
    maxmin_kernel<<<B_ROWS / ROWS_PER_BLOCK, 256, shmem, stream>>>(m, wp, out);
}